// SmoothVertices_74878459838721
// MI455X (gfx1250) — compile-verified
//
#include <hip/hip_runtime.h>

// SmoothVertices for MI455X (gfx1250).
// Memory-bound: out = copy(x) with 2 pole entries per 128x256 chart plane
// replaced by a 30-element neighbor mean. ~1 GB HBM traffic => ~43us floor
// at 23.3 TB/s. Bulk copy uses the CDNA5 async global<->LDS DMA path
// (ASYNCcnt) with 8 B128 transfers in flight per wave for latency coverage.

typedef float v4f __attribute__((ext_vector_type(4)));
typedef int   v4i __attribute__((vector_size(16)));   // matches builtin proto

#define GLOBAL_AS __attribute__((address_space(1)))
#define LDS_AS    __attribute__((address_space(3)))

#if defined(__gfx1250__) &&                                        \
    __has_builtin(__builtin_amdgcn_global_load_async_to_lds_b128) && \
    __has_builtin(__builtin_amdgcn_global_store_async_from_lds_b128) && \
    __has_builtin(__builtin_amdgcn_s_wait_asynccnt)
#define USE_ASYNC_LDS 1
#else
#define USE_ASYNC_LDS 0
#endif

// ---------------------------------------------------------------------------
// Kernel 1: pole means. One thread per (b*C+c, chart): 640 threads total.
// ws[t] = m1, ws[BC*5 + t] = m2.
// ---------------------------------------------------------------------------
__global__ void sv_reduce(const float* __restrict__ x, float* __restrict__ ws,
                          int BC) {
  int t = blockIdx.x * blockDim.x + threadIdx.x;
  if (t >= BC * 5) return;
  int ch = t % 5;
  int bc = t / 5;
  int cm = (ch + 4) % 5;                 // (c-1) mod 5
  const int HW = 128 * 256;              // 32768
  // v1 neighbors: (c,1,0) (c,1,1) (c,0,1) (c-1,127,128) (c-1,127,127)
  int o1[5] = { ch * HW + 256, ch * HW + 257, ch * HW + 1,
                cm * HW + 127 * 256 + 128, cm * HW + 127 * 256 + 127 };
  // v2 neighbors: (c,1,128) (c,1,129) (c,0,129) (c-1,127,255) (c,0,127)
  int o2[5] = { ch * HW + 384, ch * HW + 385, ch * HW + 129,
                cm * HW + 127 * 256 + 255, ch * HW + 127 };
  const long rstride = 5L * HW;          // one rotation slice
  const float* p = x + (long)bc * 6 * rstride;
  float s1 = 0.f, s2 = 0.f;
  for (int r = 0; r < 6; ++r) {
#pragma unroll
    for (int nb = 0; nb < 5; ++nb) {
      s1 += p[o1[nb]];
      s2 += p[o2[nb]];
    }
    p += rstride;
  }
  const float inv = 1.0f / 30.0f;
  ws[t]          = s1 * inv;
  ws[BC * 5 + t] = s2 * inv;
}

// ---------------------------------------------------------------------------
// Kernel 2: streaming copy. 256 threads/block, 8 x B128 per thread staged
// through LDS with async DMA (32 KB per block, no VGPR data movement).
// ---------------------------------------------------------------------------
__global__ void __launch_bounds__(256)
sv_copy(const v4f* __restrict__ in, v4f* __restrict__ out, long n4) {
  const int lane = threadIdx.x;
  const long base = (long)blockIdx.x * 2048 + lane;
#if USE_ASYNC_LDS
  __shared__ v4f buf[8][256];
#pragma unroll
  for (int k = 0; k < 8; ++k) {
    long i = base + (long)k * 256;
    if (i < n4) {
      __builtin_amdgcn_global_load_async_to_lds_b128(
          (GLOBAL_AS v4i*)(in + i), (LDS_AS v4i*)&buf[k][lane], 0, 0);
    }
  }
  __builtin_amdgcn_s_wait_asynccnt(0);
#pragma unroll
  for (int k = 0; k < 8; ++k) {
    long i = base + (long)k * 256;
    if (i < n4) {
      __builtin_amdgcn_global_store_async_from_lds_b128(
          (GLOBAL_AS v4i*)(out + i), (LDS_AS v4i*)&buf[k][lane], 0, 0);
    }
  }
  // S_ENDPGM performs an implicit wait-idle, covering outstanding stores.
#else
  v4f v[8];
#pragma unroll
  for (int k = 0; k < 8; ++k) {
    long i = base + (long)k * 256;
    if (i < n4) v[k] = __builtin_nontemporal_load(in + i);
  }
#pragma unroll
  for (int k = 0; k < 8; ++k) {
    long i = base + (long)k * 256;
    if (i < n4) __builtin_nontemporal_store(v[k], out + i);
  }
#endif
}

// ---------------------------------------------------------------------------
// Kernel 3: overwrite poles. One thread per (bc, r, chart): BC*6*5 threads.
// Flat chart-plane index == t because layout is [bc][r][chart][H*W].
// ---------------------------------------------------------------------------
__global__ void sv_fix(float* __restrict__ out, const float* __restrict__ ws,
                       int BC) {
  int t = blockIdx.x * blockDim.x + threadIdx.x;
  if (t >= BC * 30) return;
  int ch = t % 5;
  int bc = t / 30;
  long plane = (long)t * 32768;
  out[plane]       = ws[bc * 5 + ch];           // (row 0, col 0)   <- m1
  out[plane + 128] = ws[BC * 5 + bc * 5 + ch];  // (row 0, col 128) <- m2
}

// ---------------------------------------------------------------------------
extern "C" void kernel_launch(void* const* d_in, const int* in_sizes, int n_in,
                              void* d_out, int out_size, void* d_ws,
                              size_t ws_size, hipStream_t stream) {
  const float* x = (const float*)d_in[0];
  float* out = (float*)d_out;
  float* ws = (float*)d_ws;

  long n = (long)in_sizes[0];                  // 2*64*6*5*128*256 = 125,829,120
  int BC = (int)(n / (6L * 5L * 32768L));      // B*C = 128
  long n4 = n / 4;                             // float4 count = 31,457,280

  int rthreads = BC * 5;                       // 640
  sv_reduce<<<(rthreads + 255) / 256, 256, 0, stream>>>(x, ws, BC);

  long cblocks = (n4 + 2047) / 2048;           // 15,360 blocks
  sv_copy<<<(int)cblocks, 256, 0, stream>>>((const v4f*)x, (v4f*)out, n4);

  int fthreads = BC * 30;                      // 3,840
  sv_fix<<<(fthreads + 255) / 256, 256, 0, stream>>>(out, ws, BC);
}